// FlagOS_ChunkGatedDelta_11038065950742
// MI455X (gfx1250) — compile-verified
//
#include <hip/hip_runtime.h>
#include <stdint.h>

typedef __attribute__((ext_vector_type(2))) float v2f;
typedef __attribute__((ext_vector_type(8))) float v8f;

#define CHUNKC   64
#define DDIM     64
#define LD       68          // padded leading dim: breaks 64-bank aliasing on column reads
#define NTHREADS 512         // 16 waves (wave32)
#define NF4      (CHUNKC * DDIM / 4)   // 1024 float4 per 64x64 f32 tile

__device__ __forceinline__ v8f wmma_f32(v2f a, v2f b, v8f c) {
  // V_WMMA_F32_16X16X4_F32 : D = A(16x4 f32) * B(4x16 f32) + C(16x16 f32)
  return __builtin_amdgcn_wmma_f32_16x16x4_f32(
      false, a, false, b, (short)0, c, false, false);
}

// Generic LDS pointer -> 32-bit LDS offset (ISA 10.2: LDS aperture VA low 32
// bits are the logical LDS address).
__device__ __forceinline__ uint32_t lds_off32(const void* p) {
  return (uint32_t)(uintptr_t)p;
}

// GLOBAL_LOAD_ASYNC_TO_LDS_B128: 16B per lane, memory -> LDS, ASYNCcnt-tracked.
__device__ __forceinline__ void async_copy_b128(uint32_t lds_byte_off, const void* gaddr) {
  asm volatile("global_load_async_to_lds_b128 %0, %1, off"
               :: "v"(lds_byte_off), "v"((uint64_t)(uintptr_t)gaddr)
               : "memory");
}

__device__ __forceinline__ void wait_async0() {
  asm volatile("s_wait_asynccnt 0x0" ::: "memory");
}

__global__ __launch_bounds__(NTHREADS)
void FlagOS_ChunkGatedDelta_kernel(const float* __restrict__ q,
                                   const float* __restrict__ k,
                                   const float* __restrict__ v,
                                   const float* __restrict__ beta,
                                   float* __restrict__ out, int S) {
  __shared__ float sQ[2][CHUNKC * LD];
  __shared__ float sK[2][CHUNKC * LD];
  __shared__ float sV[2][CHUNKC * LD];
  __shared__ float sT[CHUNKC * LD];     // P = decay-masked Q K^T
  __shared__ float sS[DDIM * LD];       // recurrent state (lives across chunks)
  __shared__ float sCLB[CHUNKC];        // cumsum(log(beta+eps))
  __shared__ float sD2E[CHUNKC];        // exp(clb[last]-clb[c])
  __shared__ float sCD[1];              // exp(clb[last])

  const int tid  = threadIdx.x;
  const int lane = tid & 31;
  const int wave = tid >> 5;
  const int mi = wave >> 2, ni = wave & 3;      // 4x4 grid of 16x16 output tiles
  const int row0 = mi * 16, col0 = ni * 16;
  const int m   = lane & 15;                    // M (A) / N (B,C,D) in-tile index
  const int hi  = lane >> 4;
  const int kk  = hi * 2;                       // A/B sub-K offset (K0/K2 lane split)
  const int hi8 = hi * 8;                       // C/D row offset, upper lane half

  const int    bh    = blockIdx.x;
  const size_t base  = (size_t)bh * (size_t)S * DDIM;
  const size_t bbase = (size_t)bh * (size_t)S;
  const int    NC    = S / CHUNKC;

  // async double-buffer staging of one 64x64 chunk of q/k/v into padded LDS
  auto stage = [&](int cc, int buf) {
#pragma unroll
    for (int t = 0; t < NF4 / NTHREADS; ++t) {          // 2 iterations
      const int i   = tid + t * NTHREADS;               // float4 index
      const int row = i >> 4, col = (i & 15) << 2;
      const uint32_t lofs = (uint32_t)((row * LD + col) * 4);
      const size_t   goff = base + (size_t)cc * CHUNKC * DDIM + (size_t)i * 4;
      async_copy_b128(lds_off32(&sQ[buf][0]) + lofs, q + goff);
      async_copy_b128(lds_off32(&sK[buf][0]) + lofs, k + goff);
      async_copy_b128(lds_off32(&sV[buf][0]) + lofs, v + goff);
    }
  };

  for (int i = tid; i < DDIM * LD; i += NTHREADS) sS[i] = 0.f;   // zero state
  stage(0, 0);                                                    // prefetch chunk 0

  for (int c = 0; c < NC; ++c) {
    const int p = c & 1;
    wait_async0();
    __syncthreads();                     // B1: this chunk's q/k/v landed (all waves)
    if (c + 1 < NC) stage(c + 1, p ^ 1); // overlap next chunk's HBM traffic w/ compute

    // ---- gating scan: wave 0 only, shuffle-based (no serial LDS chain) ----
    if (wave == 0) {
      const float2 bb = ((const float2*)(beta + bbase + (size_t)c * CHUNKC))[lane];
      const float b0 = __logf(bb.x + 1e-6f);
      const float b1 = __logf(bb.y + 1e-6f);
      float s = b0 + b1;                       // per-pair sum
#pragma unroll
      for (int d = 1; d < 32; d <<= 1) {       // inclusive scan over 32 lanes
        float t = __shfl_up(s, d, 32);
        if (lane >= d) s += t;
      }
      const float excl = s - (b0 + b1);
      const float c0 = excl + b0;
      const float c1 = excl + b0 + b1;
      const float last = __shfl(s, 31, 32);    // clb[63]
      sCLB[2 * lane]     = c0;
      sCLB[2 * lane + 1] = c1;
      sD2E[2 * lane]     = __expf(last - c0);
      sD2E[2 * lane + 1] = __expf(last - c1);
      if (lane == 0) sCD[0] = __expf(last);
    }
    __syncthreads();                     // B2: sCLB/sD2E/sCD ready
    const float cdecay = sCD[0];

    const float* Qb = sQ[p];
    const float* Kb = sK[p];
    const float* Vb = sV[p];

    // register-cache fragments reused across matmuls
    v2f qa[16];                          // A-fragments of Q (P-phase + Q@S)
#pragma unroll
    for (int ks = 0; ks < 16; ++ks) {
      qa[ks].x = Qb[(row0 + m) * LD + 4 * ks + kk];
      qa[ks].y = Qb[(row0 + m) * LD + 4 * ks + kk + 1];
    }
    v2f vb[16];                          // B-fragments of V (P@V + state update)
#pragma unroll
    for (int ks = 0; ks < 16; ++ks) {
      vb[ks].x = Vb[(4 * ks + kk) * LD + col0 + m];
      vb[ks].y = Vb[(4 * ks + kk + 1) * LD + col0 + m];
    }

    // ---- P = (Q K^T) * decay * causal (skip strictly-upper block tiles) ----
    v8f pa = {0.f, 0.f, 0.f, 0.f, 0.f, 0.f, 0.f, 0.f};
    if (mi >= ni) {
#pragma unroll
      for (int ks = 0; ks < 16; ++ks) {
        v2f b;
        b.x = Kb[(col0 + m) * LD + 4 * ks + kk];        // B[k][n] = K[n][k]
        b.y = Kb[(col0 + m) * LD + 4 * ks + kk + 1];
        pa = wmma_f32(qa[ks], b, pa);
      }
      const int j = col0 + m;
#pragma unroll
      for (int r = 0; r < 8; ++r) {
        const int i = row0 + hi8 + r;
        const float mult = (i >= j) ? __expf(sCLB[i] - sCLB[j]) : 0.f;
        pa[r] *= mult;
      }
    }
#pragma unroll
    for (int r = 0; r < 8; ++r) sT[(row0 + hi8 + r) * LD + col0 + m] = pa[r];
    __syncthreads();                     // B3: sT ready

    // ---- O = P @ V + Q @ S_old ----
    v8f o = {0.f, 0.f, 0.f, 0.f, 0.f, 0.f, 0.f, 0.f};
#pragma unroll
    for (int ks = 0; ks < 16; ++ks) {
      v2f a;
      a.x = sT[(row0 + m) * LD + 4 * ks + kk];
      a.y = sT[(row0 + m) * LD + 4 * ks + kk + 1];
      o = wmma_f32(a, vb[ks], o);
    }
#pragma unroll
    for (int ks = 0; ks < 16; ++ks) {
      v2f b;
      b.x = sS[(4 * ks + kk) * LD + col0 + m];
      b.y = sS[(4 * ks + kk + 1) * LD + col0 + m];
      o = wmma_f32(qa[ks], b, o);
    }
    {
      float* gout = out + base + (size_t)c * CHUNKC * DDIM + row0 * DDIM + col0;
#pragma unroll
      for (int r = 0; r < 8; ++r) gout[(hi8 + r) * DDIM + m] = o[r];
    }
    __syncthreads();                     // B4: all waves done reading S_old

    // ---- S = cdecay * S + (K^T * d2e) @ V ----
    v8f u;
#pragma unroll
    for (int r = 0; r < 8; ++r)
      u[r] = sS[(row0 + hi8 + r) * LD + col0 + m] * cdecay;
#pragma unroll
    for (int ks = 0; ks < 16; ++ks) {
      v2f a;                                             // A[d][c] = K[c][d]*d2e[c]
      a.x = Kb[(4 * ks + kk) * LD + row0 + m] * sD2E[4 * ks + kk];
      a.y = Kb[(4 * ks + kk + 1) * LD + row0 + m] * sD2E[4 * ks + kk + 1];
      u = wmma_f32(a, vb[ks], u);
    }
#pragma unroll
    for (int r = 0; r < 8; ++r)
      sS[(row0 + hi8 + r) * LD + col0 + m] = u[r];
    // next iteration's B1 barrier protects sS and the LDS buffers
  }
}

extern "C" void kernel_launch(void* const* d_in, const int* in_sizes, int n_in,
                              void* d_out, int out_size, void* d_ws, size_t ws_size,
                              hipStream_t stream) {
  const float* q    = (const float*)d_in[0];
  const float* k    = (const float*)d_in[1];
  const float* v    = (const float*)d_in[2];
  const float* beta = (const float*)d_in[3];
  float* out = (float*)d_out;

  const int S  = 8192;                 // per reference
  const int BH = in_sizes[3] / S;      // beta is [B,H,S] -> B*H blocks

  hipLaunchKernelGGL(FlagOS_ChunkGatedDelta_kernel,
                     dim3(BH), dim3(NTHREADS), 0, stream,
                     q, k, v, beta, out, S);
}